// CountingDiceLoss_36146444763802
// MI455X (gfx1250) — compile-verified
//
#include <hip/hip_runtime.h>

// CountingDiceLoss for MI455X (gfx1250, wave32).
// Proven simplifications:
//  * CE term == 0 (log_softmax over a 1-wide channel axis is identically 0),
//    so the density map / centroids / bboxes are dead inputs.
//  * dice denominator 2tp+fp+fn == P + N (P = sum probs, N = class count).
// => single memory-bound streaming pass over x[:, 0:3] and y (128 MB,
//    ~5.5 us at 23.3 TB/s). Loads are non-temporal (read-once stream).

#define HW_PIX   (1024 * 1024)
#define NB       8            // batches
#define NBLK     256          // blocks per batch
#define TPB      256          // threads per block (8 wave32)
#define SMOOTH_F 1e-5f

typedef __attribute__((ext_vector_type(2))) float v2f;
typedef __attribute__((ext_vector_type(4))) float v4f;
typedef __attribute__((ext_vector_type(8))) float v8f;
typedef __attribute__((ext_vector_type(4))) int   v4i;

__device__ __forceinline__ float fast_rcp(float x) {
#if __has_builtin(__builtin_amdgcn_rcpf)
  return __builtin_amdgcn_rcpf(x);   // v_rcp_f32, ~1 ulp
#else
  return 1.0f / x;
#endif
}

// Full 32-lane sum via V_WMMA_F32_16X16X4_F32 with B = ones:
// D[m,n] = sum_k A[m,k] (row sum; independent of B's internal layout since
// every B element is 1). A layout (ISA 7.12.2): lane L<16 -> A[L][0]=v0,
// A[L][1]=v1; lane L>=16 -> A[L-16][2]=v0, A[L-16][3]=v1. With v1 = 0:
// S[m] = v[m] + v[m+16]. D layout: vgpr j holds row j (lanes 0-15) and row
// j+8 (lanes 16-31). Per-lane sum of the 8 D vgprs = half-sum; one
// shfl_xor(16) completes the 32-lane all-reduce, full f32 precision.
__device__ __forceinline__ float wave_sum32(float v) {
#if __has_builtin(__builtin_amdgcn_wmma_f32_16x16x4_f32)
  v2f a;  a[0] = v;     a[1] = 0.0f;
  v2f bo; bo[0] = 1.0f; bo[1] = 1.0f;
  v8f c = {0.f, 0.f, 0.f, 0.f, 0.f, 0.f, 0.f, 0.f};
  v8f d = __builtin_amdgcn_wmma_f32_16x16x4_f32(
      /*neg_a=*/false, a, /*neg_b=*/false, bo,
      /*c_mod=*/(short)0, c, /*reuse_a=*/false, /*reuse_b=*/false);
  float h = ((d[0] + d[1]) + (d[2] + d[3])) + ((d[4] + d[5]) + (d[6] + d[7]));
  h += __shfl_xor(h, 16, 32);
  return h;
#else
  #pragma unroll
  for (int off = 16; off > 0; off >>= 1) v += __shfl_xor(v, off, 32);
  return v;
#endif
}

// Kernel 1: streaming 3-way softmax + per-block partial sums.
// Record per (batch, block): {P0,P1,P2, TP0,TP1,TP2, N0,N1,N2}
__global__ __launch_bounds__(TPB) void dice_partials_k(
    const float* __restrict__ x, const int* __restrict__ y,
    float* __restrict__ ws) {
  const int b   = blockIdx.y;
  const int blk = blockIdx.x;
  const int tid = threadIdx.x;

  const v4f* xc0 = (const v4f*)(x + (size_t)(b * 4 + 0) * HW_PIX);
  const v4f* xc1 = (const v4f*)(x + (size_t)(b * 4 + 1) * HW_PIX);
  const v4f* xc2 = (const v4f*)(x + (size_t)(b * 4 + 2) * HW_PIX);
  const v4i* yl  = (const v4i*)(y + (size_t)b * HW_PIX);

  float P0 = 0.f, P1 = 0.f, P2 = 0.f;
  float T0 = 0.f, T1 = 0.f, T2 = 0.f;
  float N0 = 0.f, N1 = 0.f, N2 = 0.f;

  const int vecPerBlk = (HW_PIX / 4) / NBLK;  // 1024 x v4f per block
  const int iters     = vecPerBlk / TPB;      // 4 per thread
  int vi = blk * vecPerBlk + tid;
  #pragma unroll
  for (int it = 0; it < iters; ++it, vi += TPB) {
    const v4f a  = __builtin_nontemporal_load(xc0 + vi);  // TH=NT streams
    const v4f bb = __builtin_nontemporal_load(xc1 + vi);
    const v4f cc = __builtin_nontemporal_load(xc2 + vi);
    const v4i lv = __builtin_nontemporal_load(yl + vi);
    #pragma unroll
    for (int e = 0; e < 4; ++e) {
      const float xa = a[e], xb = bb[e], xc = cc[e];
      const float m  = fmaxf(fmaxf(xa, xb), xc);
      const float e0 = __expf(xa - m);
      const float e1 = __expf(xb - m);
      const float e2 = __expf(xc - m);
      const float r  = fast_rcp(e0 + e1 + e2);  // v_rcp_f32, no IEEE div seq
      const float p0 = e0 * r, p1 = e1 * r, p2 = e2 * r;
      P0 += p0; P1 += p1; P2 += p2;
      const int k = lv[e];
      T0 += (k == 0) ? p0 : 0.0f;
      T1 += (k == 1) ? p1 : 0.0f;
      T2 += (k == 2) ? p2 : 0.0f;
      N0 += (k == 0) ? 1.0f : 0.0f;
      N1 += (k == 1) ? 1.0f : 0.0f;
      N2 += (k == 2) ? 1.0f : 0.0f;
    }
  }

  // Wave-level all-reduce via WMMA (EXEC all-ones here: no divergence yet).
  float vals[9] = {P0, P1, P2, T0, T1, T2, N0, N1, N2};
  float tot[9];
  #pragma unroll
  for (int i = 0; i < 9; ++i) tot[i] = wave_sum32(vals[i]);

  __shared__ float red[(TPB / 32) * 9];
  const int wv = tid >> 5, ln = tid & 31;
  if (ln == 0) {
    #pragma unroll
    for (int i = 0; i < 9; ++i) red[wv * 9 + i] = tot[i];
  }
  __syncthreads();
  if (tid < 9) {
    float s = 0.f;
    #pragma unroll
    for (int w = 0; w < TPB / 32; ++w) s += red[w * 9 + tid];
    ws[((size_t)b * NBLK + blk) * 9 + tid] = s;
  }
}

// Kernel 2: one block; wave w reduces batch w's NBLK records, computes dice.
__global__ __launch_bounds__(256) void dice_finalize_k(
    const float* __restrict__ ws, float* __restrict__ out) {
  const int tid  = threadIdx.x;
  const int b    = tid >> 5;   // wave index == batch (8 waves, B == 8)
  const int lane = tid & 31;

  float acc[9] = {0.f, 0.f, 0.f, 0.f, 0.f, 0.f, 0.f, 0.f, 0.f};
  for (int j = lane; j < NBLK; j += 32) {
    const float* r = ws + ((size_t)b * NBLK + j) * 9;
    #pragma unroll
    for (int i = 0; i < 9; ++i) acc[i] += r[i];
  }

  float tot[9];
  #pragma unroll
  for (int i = 0; i < 9; ++i) tot[i] = wave_sum32(acc[i]);

  __shared__ float contrib[NB];
  if (lane == 0) {
    // Order: P0 P1 P2 | TP0 TP1 TP2 | N0 N1 N2 ; do_bg drops class 0.
    const float dc1 = (2.0f * tot[4] + SMOOTH_F) / (tot[1] + tot[7] + SMOOTH_F);
    const float dc2 = (2.0f * tot[5] + SMOOTH_F) / (tot[2] + tot[8] + SMOOTH_F);
    contrib[b] = dc1 + dc2;
  }
  __syncthreads();
  if (tid == 0) {
    float s = 0.f;
    #pragma unroll
    for (int w = 0; w < NB; ++w) s += contrib[w];
    out[0] = -(s / 16.0f);  // + CE, which is exactly 0
  }
}

extern "C" void kernel_launch(void* const* d_in, const int* in_sizes, int n_in,
                              void* d_out, int out_size, void* d_ws, size_t ws_size,
                              hipStream_t stream) {
  const float* x = (const float*)d_in[0];  // [8,4,1024,1024] f32
  const int*   y = (const int*)d_in[1];    // [8,1,1024,1024] i32
  // d_in[2..4] (cent_i, cent_j, bbox) are provably dead inputs.
  float* ws  = (float*)d_ws;               // needs 8*256*9*4 = 73728 bytes
  float* out = (float*)d_out;

  dice_partials_k<<<dim3(NBLK, NB), TPB, 0, stream>>>(x, y, ws);
  dice_finalize_k<<<1, 256, 0, stream>>>(ws, out);
}